// ConvKNRMLayer_60885456388505
// MI455X (gfx1250) — compile-verified
//
#include <hip/hip_runtime.h>
#include <hip/hip_bf16.h>

// ---------------------------------------------------------------------------
// Conv-KNRM for MI455X (gfx1250, wave32).
// All GEMMs run on V_WMMA_F32_16X16X4_F32 to preserve full fp32 sim accuracy
// (the sigma=0.001 pooling bin demands it). Doc features (96 MB fp32) live in
// workspace and stay resident in the 192 MB L2 between stages.
// ---------------------------------------------------------------------------

#define B_   128
#define QL_  16
#define DL_  512
#define DM_  256
#define NF_  128
#define NB_  11

typedef float v2f __attribute__((ext_vector_type(2)));
typedef float v8f __attribute__((ext_vector_type(8)));

__constant__ float c_mus[NB_] = {1.0f, 0.9f, 0.7f, 0.5f, 0.3f, 0.1f,
                                 -0.1f, -0.3f, -0.5f, -0.7f, -0.9f};
// -1/(2*sigma^2):  sigma = 0.001 for bin 0, 0.1 for the rest
__constant__ float c_n2s[NB_] = {-500000.0f, -50.0f, -50.0f, -50.0f, -50.0f,
                                 -50.0f, -50.0f, -50.0f, -50.0f, -50.0f, -50.0f};
// pair order: uu, ut, ub, bu, tu, bb, bt, tb, tt   (g: 0=uni,1=bi,2=tri)
__constant__ int c_qg[9] = {0, 0, 0, 1, 2, 1, 1, 2, 2};
__constant__ int c_dg[9] = {0, 2, 1, 0, 0, 1, 2, 1, 2};

// ---------------------------------------------------------------------------
// Stage 1: n-gram conv features + bias + ReLU + L2-norm over NF.
//   x   : [B, T, DM]
//   out : [B, 3, T, NF]  (L2-normalized over NF)
// One block per (batch, 16-position tile); 256 threads = 8 waves; each wave
// owns a 16-row NF tile (nfb = wave*16). Conv shifts are folded into the
// B-fragment row selection from the 18-row LDS halo tile.
// ---------------------------------------------------------------------------
__global__ __launch_bounds__(256)
void conv_feat_kernel(const float* __restrict__ x,
                      const float* __restrict__ wu, const float* __restrict__ bu,
                      const float* __restrict__ wb, const float* __restrict__ bbi,
                      const float* __restrict__ wt, const float* __restrict__ bt,
                      float* __restrict__ out, int T, int tilesPerB)
{
    __shared__ float xs[18 * DM_];        // positions t0-1 .. t0+16
    __shared__ float ssq[3][16];          // sum of squares per (group, position)

    const int b    = blockIdx.x / tilesPerB;
    const int t0   = (blockIdx.x % tilesPerB) * 16;
    const int tid  = threadIdx.x;
    const int wave = tid >> 5;
    const int lane = tid & 31;
    const int hi   = lane >> 4;           // selects K pair (A/B) and M+8 (C)
    const int mn   = lane & 15;           // M row for A, N col for B/C
    const int nfb  = wave * 16;           // this wave's NF tile base

    // stage halo tile of embeddings
    for (int idx = tid; idx < 18 * DM_; idx += 256) {
        int row = idx / DM_;
        int dm  = idx % DM_;
        int p   = t0 - 1 + row;
        xs[idx] = (p >= 0 && p < T) ? x[((size_t)b * T + p) * DM_ + dm] : 0.0f;
    }
    if (tid < 48) ((float*)ssq)[tid] = 0.0f;
    __syncthreads();

    const float* wptr[3]  = {wu, wb, wt};
    const float* bias[3]  = {bu, bbi, bt};
    const int    width[3] = {1, 2, 3};
    const int    rbase[3] = {1, 0, 0};    // uni reads x[t]; bi/tri start at x[t-1]

#pragma unroll
    for (int g = 0; g < 3; ++g) {
        const int W    = width[g];
        const int Kdim = DM_ * W;
        const float* Wm = wptr[g];

        v8f acc = {0.f, 0.f, 0.f, 0.f, 0.f, 0.f, 0.f, 0.f};
        for (int k0 = 0; k0 < Kdim; k0 += 4) {
            const int kk  = k0 + 2 * hi;          // this lane's K pair base
            const int dm  = kk & (DM_ - 1);
            const int tap = kk >> 8;              // DM_ == 256
            // A: weights, im2col K = tap*DM + dm  -> w[nf][dm][tap]
            const float* wr = Wm + ((size_t)(nfb + mn) * DM_ + dm) * W + tap;
            v2f a; a[0] = wr[0]; a[1] = wr[W];
            // B: embeddings from LDS halo; k+1 -> dm+1, same tap/row
            const int row = mn + tap + rbase[g];
            v2f bv; bv[0] = xs[row * DM_ + dm]; bv[1] = xs[row * DM_ + dm + 1];
            acc = __builtin_amdgcn_wmma_f32_16x16x4_f32(
                false, a, false, bv, (short)0, acc, false, false);
        }

        // bias + ReLU, partial sum-of-squares over this lane's 8 NF rows
        float vals[8];
        float part = 0.0f;
#pragma unroll
        for (int r = 0; r < 8; ++r) {
            float v = acc[r] + bias[g][nfb + 8 * hi + r];
            v = fmaxf(v, 0.0f);
            vals[r] = v;
            part += v * v;
        }
        atomicAdd(&ssq[g][mn], part);             // ds_add_f32
        __syncthreads();

        const float inv = 1.0f / fmaxf(sqrtf(ssq[g][mn]), 1e-10f);
        float* op = out + (((size_t)b * 3 + g) * T + (t0 + mn)) * NF_ + nfb + 8 * hi;
        float4 lo = make_float4(vals[0] * inv, vals[1] * inv, vals[2] * inv, vals[3] * inv);
        float4 hi4 = make_float4(vals[4] * inv, vals[5] * inv, vals[6] * inv, vals[7] * inv);
        ((float4*)op)[0] = lo;                    // global_store_b128
        ((float4*)(op + 4))[0] = hi4;
    }
}

// ---------------------------------------------------------------------------
// Stage 2: cosine-sim GEMM (16 x 512, K = 128) fused with Gaussian kernel
// pooling. One block (4 waves) per (batch, pair); each wave handles 8 of the
// 32 d-tiles; per-lane register accumulators (fixed q-rows, fixed d mod 16)
// are merged once at the end through LDS atomics.
//   pooled : [9, B, NB]
// ---------------------------------------------------------------------------
__global__ __launch_bounds__(128)
void sim_pool_kernel(const float* __restrict__ qf,   // [B,3,QL,NF]
                     const float* __restrict__ df,   // [B,3,DL,NF]
                     const float* __restrict__ qm,   // [B,QL]
                     const float* __restrict__ dmk,  // [B,DL]
                     float* __restrict__ pooled)
{
    __shared__ float qs[QL_ * NF_];       // 8 KB query-feature tile
    __shared__ float sAcc[QL_][NB_];
    __shared__ float lAcc[QL_][NB_];

    const int p    = blockIdx.x % 9;
    const int b    = blockIdx.x / 9;
    const int tid  = threadIdx.x;
    const int wave = tid >> 5;
    const int lane = tid & 31;
    const int hi   = lane >> 4;
    const int nm   = lane & 15;

    const int qg = c_qg[p], dg = c_dg[p];
    const float* qbase = qf + ((size_t)b * 3 + qg) * QL_ * NF_;
    const float* dbase = df + ((size_t)b * 3 + dg) * DL_ * NF_;

    for (int idx = tid; idx < QL_ * NF_; idx += 128) qs[idx] = qbase[idx];
    if (tid < QL_ * NB_) ((float*)sAcc)[tid] = 0.0f;
    __syncthreads();

    float pa[8][NB_];
#pragma unroll
    for (int r = 0; r < 8; ++r)
#pragma unroll
        for (int j = 0; j < NB_; ++j) pa[r][j] = 0.0f;

    const float* qrow = qs + nm * NF_ + 2 * hi;   // A-fragment source (LDS)

    for (int it = 0; it < 8; ++it) {
        const int d  = (wave * 8 + it) * 16 + nm;
        const float md = dmk[b * DL_ + d];
        const float* bcol = dbase + (size_t)d * NF_ + 2 * hi;
        if (it + 1 < 8)                            // stream-ahead: global_prefetch_b8
            __builtin_prefetch(bcol + 16 * NF_, 0, 3);

        v8f sim = {0.f, 0.f, 0.f, 0.f, 0.f, 0.f, 0.f, 0.f};
        for (int k0 = 0; k0 < NF_; k0 += 4) {
            v2f a;  a[0]  = qrow[k0]; a[1]  = qrow[k0 + 1];   // ds_load_b64
            v2f bv; bv[0] = bcol[k0]; bv[1] = bcol[k0 + 1];   // global_load_b64
            sim = __builtin_amdgcn_wmma_f32_16x16x4_f32(
                false, a, false, bv, (short)0, sim, false, false);
        }
        // Gaussian bins, masked by d_mask; accumulate over D in registers
#pragma unroll
        for (int r = 0; r < 8; ++r) {
            const float s = sim[r];
#pragma unroll
            for (int j = 0; j < NB_; ++j) {
                const float dlt = s - c_mus[j];
                pa[r][j] += __expf(dlt * dlt * c_n2s[j]) * md;
            }
        }
    }

    // merge per-lane partials: q = r + 8*hi
#pragma unroll
    for (int r = 0; r < 8; ++r)
#pragma unroll
        for (int j = 0; j < NB_; ++j)
            atomicAdd(&sAcc[r + 8 * hi][j], pa[r][j]);
    __syncthreads();

    if (tid < QL_ * NB_) {
        const int q = tid / NB_, j = tid % NB_;
        const float sv = fmaxf(sAcc[q][j], 1e-10f);
        lAcc[q][j] = __logf(sv) * 0.01f * qm[b * QL_ + q];
    }
    __syncthreads();

    if (tid < NB_) {
        float t = 0.0f;
#pragma unroll
        for (int q = 0; q < QL_; ++q) t += lAcc[q][tid];
        pooled[((size_t)p * B_ + b) * NB_ + tid] = t;
    }
}

// ---------------------------------------------------------------------------
// Stage 3: dense [B,99] @ w_dense + bias -> tanh -> [B]
// ---------------------------------------------------------------------------
__global__ __launch_bounds__(128)
void dense_tanh_kernel(const float* __restrict__ pooled,
                       const float* __restrict__ wd,
                       const float* __restrict__ bd,
                       float* __restrict__ outp)
{
    const int b = threadIdx.x;
    if (b >= B_) return;
    float acc = bd[0];
#pragma unroll
    for (int j = 0; j < 9 * NB_; ++j) {
        const int pp = j / NB_, bin = j % NB_;
        acc += pooled[((size_t)pp * B_ + b) * NB_ + bin] * wd[j];
    }
    outp[b] = tanhf(acc);
}

// ---------------------------------------------------------------------------
extern "C" void kernel_launch(void* const* d_in, const int* in_sizes, int n_in,
                              void* d_out, int out_size, void* d_ws, size_t ws_size,
                              hipStream_t stream)
{
    const float* query  = (const float*)d_in[0];
    const float* cand   = (const float*)d_in[1];
    const float* q_mask = (const float*)d_in[2];
    const float* d_mask = (const float*)d_in[3];
    const float* w_uni  = (const float*)d_in[4];
    const float* b_uni  = (const float*)d_in[5];
    const float* w_bi   = (const float*)d_in[6];
    const float* b_bi   = (const float*)d_in[7];
    const float* w_tri  = (const float*)d_in[8];
    const float* b_tri  = (const float*)d_in[9];
    const float* w_dense = (const float*)d_in[10];
    const float* b_dense = (const float*)d_in[11];
    float* outp = (float*)d_out;

    // workspace carve-out (~104 MB): doc feats | query feats | pooled
    const size_t dfeatElems = (size_t)B_ * 3 * DL_ * NF_;   // 25,165,824
    const size_t qfeatElems = (size_t)B_ * 3 * QL_ * NF_;   //    786,432
    float* dfeat  = (float*)d_ws;
    float* qfeat  = dfeat + dfeatElems;
    float* pooled = qfeat + qfeatElems;

    // doc features: 128 batches x 32 tiles
    conv_feat_kernel<<<B_ * (DL_ / 16), 256, 0, stream>>>(
        cand, w_uni, b_uni, w_bi, b_bi, w_tri, b_tri, dfeat, DL_, DL_ / 16);
    // query features: 128 batches x 1 tile
    conv_feat_kernel<<<B_, 256, 0, stream>>>(
        query, w_uni, b_uni, w_bi, b_bi, w_tri, b_tri, qfeat, QL_, 1);
    // 9 sim GEMMs + kernel pooling
    sim_pool_kernel<<<B_ * 9, 128, 0, stream>>>(qfeat, dfeat, q_mask, d_mask, pooled);
    // final dense + tanh
    dense_tanh_kernel<<<1, 128, 0, stream>>>(pooled, w_dense, b_dense, outp);
}